// LSTM_CELL_68496138436629
// MI455X (gfx1250) — compile-verified
//
#include <hip/hip_runtime.h>
#include <math.h>

typedef float v2f __attribute__((ext_vector_type(2)));
typedef float v8f __attribute__((ext_vector_type(8)));

#define STATE   65536
#define INPUT   65536
#define CONCAT  (STATE + INPUT)
#define H1V     100
#define H2V     50
#define NG      4
#define ROWS1   (NG * H1V)          // 400 = 25 * 16
#define ROWT1   (ROWS1 / 16)        // 25 row tiles
#define KCHUNK  2048                // divides 65536 -> chunk never straddles concat boundary
#define NKCH    (CONCAT / KCHUNK)   // 64
#define KSPAN   (KCHUNK / 8)        // 256 K per wave

__device__ __forceinline__ float sigmoidf_(float v) { return 1.0f / (1.0f + expf(-v)); }

__device__ __forceinline__ v8f wmma_f32_16x16x4(v2f a, v2f b, v8f c) {
    return __builtin_amdgcn_wmma_f32_16x16x4_f32(
        false, a, false, b, (short)0, c, false, false);
}

// ---------------------------------------------------------------------------
// 0) zero the split-K accumulator for h1
// ---------------------------------------------------------------------------
__global__ void zero_ws_kernel(float* __restrict__ p, int n) {
    int i = blockIdx.x * blockDim.x + threadIdx.x;
    if (i < n) p[i] = 0.0f;
}

// ---------------------------------------------------------------------------
// 1) h1 = W1 @ cat(hyp, x)   via V_WMMA_F32_16X16X4_F32, split-K + atomics
//    grid = 25 row-tiles * 64 K-chunks; block = 256 (8 waves)
//    Constant trip count + dual accumulators -> scalar loop control,
//    paired loads in flight before each wmma pair.
// ---------------------------------------------------------------------------
__global__ __launch_bounds__(256) void gemv1_wmma(
    const float* __restrict__ W1, const float* __restrict__ hyp,
    const float* __restrict__ x, float* __restrict__ h1sum)
{
    __shared__ float partials[16];
    const int tid   = threadIdx.x;
    const int lane  = tid & 31;
    const int wave  = tid >> 5;
    const int rowT  = blockIdx.x / NKCH;
    const int kCh   = blockIdx.x % NKCH;
    const int kBase = kCh * KCHUNK;

    // chunk lies entirely inside hypotesis or x
    const float* src = (kBase < STATE) ? hyp : x;

    const int m     = lane & 15;          // matrix row within tile (A layout)
    const int hiOff = (lane >> 4) << 1;   // lanes 16-31 supply K=+2,+3

    const int kw = kBase + wave * KSPAN;
    const float* aPtr = W1 + (size_t)(rowT * 16 + m) * CONCAT + kw + hiOff;
    const float* bPtr = src + ((kw + hiOff) & (STATE - 1));

    if (tid < 16) partials[tid] = 0.0f;
    __syncthreads();

    v8f c0 = {}, c1 = {};
    #pragma unroll 4
    for (int i = 0; i < KSPAN / 8; ++i) {       // constant trip count = 32
        v2f a0 = *(const v2f*)(aPtr + 8 * i);
        v2f b0 = *(const v2f*)(bPtr + 8 * i);
        v2f a1 = *(const v2f*)(aPtr + 8 * i + 4);
        v2f b1 = *(const v2f*)(bPtr + 8 * i + 4);
        c0 = wmma_f32_16x16x4(a0, b0, c0);
        c1 = wmma_f32_16x16x4(a1, b1, c1);
    }
    c0 += c1;

    // D layout: lanes 0-15 hold rows 0-7 in c[0..7]; lanes 16-31 rows 8-15.
    // All N columns are identical (B broadcast) -> lanes 0 and 16 suffice.
    if ((lane & 15) == 0) {
        const int half = lane >> 4;
        #pragma unroll
        for (int r = 0; r < 8; ++r)
            atomicAdd(&partials[half * 8 + r], c0[r]);      // ds_add_f32
    }
    __syncthreads();
    if (tid < 16) atomicAdd(&h1sum[rowT * 16 + tid], partials[tid]);
}

// ---------------------------------------------------------------------------
// 2) h2 = W2 @ (h1 + b1) + b2      (tiny: 20K weights, one block)
// ---------------------------------------------------------------------------
__global__ __launch_bounds__(256) void mlp2_kernel(
    const float* __restrict__ h1sum, const float* __restrict__ b1,
    const float* __restrict__ W2, const float* __restrict__ b2,
    float* __restrict__ h2out)
{
    __shared__ float h1s[ROWS1];
    const int t = threadIdx.x;
    for (int i = t; i < ROWS1; i += blockDim.x) h1s[i] = h1sum[i] + b1[i];
    __syncthreads();
    if (t < NG * H2V) {
        const int g = t / H2V;
        float sum = b2[t];
        const float* w  = W2 + (size_t)t * H1V;     // W2[g][o][:] contiguous
        const float* hv = h1s + g * H1V;
        #pragma unroll 4
        for (int h = 0; h < H1V; ++h) sum += w[h] * hv[h];
        h2out[t] = sum;
    }
}

// ---------------------------------------------------------------------------
// 3) g = W3 @ h2 + b3, fused gate math -> (new_hypotesis, new_cell_state)
//    grid = STATE/128; block = 256 (8 waves, 16 states each)
// ---------------------------------------------------------------------------
__global__ __launch_bounds__(256) void gemv3_wmma_fused(
    const float* __restrict__ W3, const float* __restrict__ b3,
    const float* __restrict__ h2, const float* __restrict__ cprev,
    float* __restrict__ out)
{
    __shared__ float h2s[NG * H2V];
    const int tid = threadIdx.x;
    if (tid < NG * H2V) h2s[tid] = h2[tid];
    __syncthreads();

    const int lane  = tid & 31;
    const int wave  = tid >> 5;
    const int sBase = blockIdx.x * 128 + wave * 16;
    const int m     = lane & 15;
    const int hiOff = (lane >> 4) << 1;

    v8f d[NG];
    #pragma unroll
    for (int g = 0; g < NG; ++g) {
        const float* rowPtr = W3 + ((size_t)g * STATE + (sBase + m)) * H2V;
        const float* hvec   = h2s + g * H2V;
        v8f c = {};
        // K = 50, padded to 52 with select-based masking (no divergent CF,
        // EXEC stays all-ones for WMMA; clamped index keeps loads in-bounds)
        #pragma unroll
        for (int k0 = 0; k0 < 52; k0 += 4) {
            const int   ka    = k0 + hiOff;
            const bool  ok    = (ka < H2V);
            const int   kc    = ok ? ka : 0;
            const float vmask = ok ? 1.0f : 0.0f;
            v2f a = *(const v2f*)(rowPtr + kc);
            v2f b = *(const v2f*)(hvec + kc);
            a *= vmask;
            b *= vmask;
            c = wmma_f32_16x16x4(a, b, c);
        }
        d[g] = c;
    }

    // lanes 0/16 own the 16 rows -> fused elementwise gate epilogue
    if ((lane & 15) == 0) {
        const int half = lane >> 4;
        #pragma unroll
        for (int r = 0; r < 8; ++r) {
            const int s = sBase + half * 8 + r;
            const float gf = d[0][r] + b3[0 * STATE + s];
            const float gi = d[1][r] + b3[1 * STATE + s];
            const float gc = d[2][r] + b3[2 * STATE + s];
            const float go = d[3][r] + b3[3 * STATE + s];
            const float f    = sigmoidf_(gf);
            const float ig   = sigmoidf_(gi);
            const float chat = tanhf(gc);
            const float og   = sigmoidf_(go);
            const float nc   = f * cprev[s] + ig * chat;
            const float nh   = tanhf(nc) * og;
            out[s]         = nh;   // new_hypotesis
            out[STATE + s] = nc;   // new_cell_state
        }
    }
}

// ---------------------------------------------------------------------------
extern "C" void kernel_launch(void* const* d_in, const int* in_sizes, int n_in,
                              void* d_out, int out_size, void* d_ws, size_t ws_size,
                              hipStream_t stream) {
    const float* cprev = (const float*)d_in[0];
    const float* hyp   = (const float*)d_in[1];
    const float* x     = (const float*)d_in[2];
    const float* W1    = (const float*)d_in[3];
    const float* b1    = (const float*)d_in[4];
    const float* W2    = (const float*)d_in[5];
    const float* b2    = (const float*)d_in[6];
    const float* W3    = (const float*)d_in[7];
    const float* b3    = (const float*)d_in[8];
    float* out = (float*)d_out;
    float* ws  = (float*)d_ws;

    float* h1sum = ws;           // 400 floats (atomically accumulated)
    float* h2buf = ws + ROWS1;   // 200 floats

    zero_ws_kernel<<<2, 256, 0, stream>>>(h1sum, ROWS1);
    gemv1_wmma<<<ROWT1 * NKCH, 256, 0, stream>>>(W1, hyp, x, h1sum);
    mlp2_kernel<<<1, 256, 0, stream>>>(h1sum, b1, W2, b2, h2buf);
    gemv3_wmma_fused<<<STATE / 128, 256, 0, stream>>>(W3, b3, h2buf, cprev, out);
}